// MBPGNN_27642409517716
// MI455X (gfx1250) — compile-verified
//
#include <hip/hip_runtime.h>
#include <math.h>
#include <stdint.h>

#define F_DIM 400
#define BGRAPH 16
#define NNODE (BGRAPH * F_DIM)   // 6400
#define NEDGE 204800
#define CH 64
#define SCH 256
#define HMID 512
#define NCLS 7
#define TRIU_N 79800
#define D1_N (TRIU_N + 4 * SCH)  // 80824

typedef float v2f __attribute__((ext_vector_type(2)));
typedef float v8f __attribute__((ext_vector_type(8)));

// ---------------------------------------------------------------------------
// CDNA5 async global->LDS copies (ASYNCcnt-tracked, GV addressing mode).
// Inline asm per CDNA5 ISA / 15.18.3 (portable across
// ROCm7.2 and amdgpu-toolchain, unlike the TDM builtin).
// ---------------------------------------------------------------------------
__device__ __forceinline__ void async_g2l_b128(uint32_t lds_addr, const void* gaddr)
{
    asm volatile("global_load_async_to_lds_b128 %0, %1, off"
                 :: "v"(lds_addr), "v"((uint64_t)gaddr) : "memory");
}
__device__ __forceinline__ void async_g2l_b32(uint32_t lds_addr, const void* gaddr)
{
    asm volatile("global_load_async_to_lds_b32 %0, %1, off"
                 :: "v"(lds_addr), "v"((uint64_t)gaddr) : "memory");
}
__device__ __forceinline__ void wait_async0()
{
    asm volatile("s_wait_asynccnt 0x0" ::: "memory");
}

// ---------------------------------------------------------------------------
// Generic WMMA f32 GEMM: C[MxN] = act(A[MxK] @ B[KxN] (+bias))
// flags: bit0 = relu, bit1 = accumulate into C
// B column-tile (K x 16) staged in LDS via async global->LDS B128 copies,
// shared by all waves of the block. One wave computes one 16x16 tile via
// V_WMMA_F32_16X16X4_F32 over K/4 steps.
// Requires: K % 4 == 0, K <= 512, N % 16 == 0, M % 16 == 0.
// ---------------------------------------------------------------------------
#define GK_MAXK 512
__global__ void k_wmma_gemm(const float* __restrict__ A, const float* __restrict__ Bm,
                            const float* __restrict__ bias, float* __restrict__ Cm,
                            int M, int Nn, int K, int flags)
{
    __shared__ __attribute__((aligned(16))) float ldsB[GK_MAXK * 16];
    const int lane = threadIdx.x & 31;
    const int wave = threadIdx.x >> 5;
    const int col0 = blockIdx.y * 16;

    // stage B[:, col0:col0+16] into LDS: one b128 async op per 4 floats
    const int quads = K * 4;                      // (K*16)/4 elements of 16B
    for (int q4 = threadIdx.x; q4 < quads; q4 += blockDim.x) {
        int k = q4 >> 2, nq = (q4 & 3) * 4;
        async_g2l_b128((uint32_t)(uintptr_t)&ldsB[k * 16 + nq],
                       Bm + (size_t)k * Nn + col0 + nq);
    }
    wait_async0();
    __syncthreads();

    const int row0 = (blockIdx.x * (blockDim.x >> 5) + wave) * 16;
    if (row0 >= M) return;

    const int half = lane >> 4;   // 0: K=k0..k0+1 , 1: K=k0+2..k0+3
    const int mn   = lane & 15;   // A: row index, B: col index
    v8f acc = {};
    const float* Arow = A + (size_t)(row0 + mn) * K + 2 * half;
    for (int k0 = 0; k0 < K; k0 += 4) {
        v2f a = *(const v2f*)(Arow + k0);              // global_load_b64
        v2f b;
        b.x = ldsB[(k0 + 2 * half) * 16 + mn];         // ds_load (conflict-free)
        b.y = ldsB[(k0 + 2 * half + 1) * 16 + mn];
        acc = __builtin_amdgcn_wmma_f32_16x16x4_f32(false, a, false, b,
                                                    (short)0, acc, false, false);
    }
    const float bv = bias ? bias[col0 + mn] : 0.f;
    #pragma unroll
    for (int j = 0; j < 8; ++j) {
        int m = row0 + j + 8 * half;                   // documented C/D striping
        float val = acc[j] + bv;
        if (flags & 1) val = fmaxf(val, 0.f);
        size_t o = (size_t)m * Nn + col0 + mn;
        if (flags & 2) Cm[o] += val; else Cm[o] = val;
    }
}

// ---------------------------------------------------------------------------
// Spherical harmonics per edge (16 components)
// ---------------------------------------------------------------------------
__global__ void k_sph(const float* __restrict__ ev, float* __restrict__ Y)
{
    int e = blockIdx.x * blockDim.x + threadIdx.x;
    if (e >= NEDGE) return;
    float x = ev[3*e], y = ev[3*e+1], z = ev[3*e+2];
    float inv = rsqrtf(x*x + y*y + z*z + 1e-9f);
    x *= inv; y *= inv; z *= inv;
    float z2 = z*z;
    float c[16] = {1.f, x, y, z, x*y, y*z, 3.f*z2 - 1.f, z*x, x*x - y*y,
                   y*(3.f*x*x - y*y), x*y*z, y*(5.f*z2 - 1.f), z*(5.f*z2 - 3.f),
                   x*(5.f*z2 - 1.f), z*(x*x - y*y), x*(x*x - 3.f*y*y)};
    #pragma unroll
    for (int l = 0; l < 16; ++l) Y[(size_t)e*16 + l] = c[l];
}

// ---------------------------------------------------------------------------
// Edge messages + scatter-add into A[n][c][l]; /AVG_NEIGH folded in (0.1)
// A (26 MB) is L2-resident -> f32 atomics resolve in L2.
// ---------------------------------------------------------------------------
__global__ void k_edge(const float* __restrict__ h, const float* __restrict__ ea,
                       const float* __restrict__ Wrad, const float* __restrict__ brad,
                       const int* __restrict__ src, const int* __restrict__ dst,
                       const float* __restrict__ Y, float* __restrict__ Abuf)
{
    int t = blockIdx.x * blockDim.x + threadIdx.x;
    if (t >= NEDGE * CH) return;
    int e = t >> 6, c = t & 63;
    float w  = ea[e] * Wrad[c] + brad[c];
    float hc = h[(size_t)src[e] * CH + c] * w * 0.1f;
    const float* Ye = Y + (size_t)e * 16;
    float* Ad = Abuf + ((size_t)dst[e] * CH + c) * 16;
    #pragma unroll
    for (int l = 0; l < 16; ++l) atomicAdd(&Ad[l], hc * Ye[l]);
}

// ---------------------------------------------------------------------------
// Per-node invariants: S=[s|s2|s3] (Nx192), V2_i=[v_i | s*v_i] (Nx128, i=0..2)
// ---------------------------------------------------------------------------
__global__ void k_build_S(const float* __restrict__ Abuf, float* __restrict__ S,
                          float* __restrict__ V2)
{
    int t = blockIdx.x * blockDim.x + threadIdx.x;
    if (t >= NNODE * CH) return;
    int n = t >> 6, c = t & 63;
    const float* a = Abuf + ((size_t)n * CH + c) * 16;
    float s = a[0], s2 = 0.f;
    #pragma unroll
    for (int l = 0; l < 16; ++l) s2 += a[l] * a[l];
    float s3 = s * s2;
    S[(size_t)n*192 +       c] = s;
    S[(size_t)n*192 +  64 + c] = s2;
    S[(size_t)n*192 + 128 + c] = s3;
    #pragma unroll
    for (int i = 0; i < 3; ++i) {
        float vi = a[1 + i];
        V2[(size_t)i*NNODE*128 + (size_t)n*128 +      c] = vi;
        V2[(size_t)i*NNODE*128 + (size_t)n*128 + 64 + c] = s * vi;
    }
}

// nf[n] = tanh([scal(256) | vec(o,i) flattened o*3+i (768)])
__global__ void k_finalize_nf(const float* __restrict__ scal, const float* __restrict__ vecb,
                              float* __restrict__ nf)
{
    int t = blockIdx.x * blockDim.x + threadIdx.x;
    if (t >= NNODE * SCH) return;
    int n = t >> 8, o = t & 255;
    nf[(size_t)n*1024 + o] = tanhf(scal[(size_t)n*SCH + o]);
    #pragma unroll
    for (int i = 0; i < 3; ++i)
        nf[(size_t)n*1024 + 256 + o*3 + i] =
            tanhf(vecb[(size_t)i*NNODE*SCH + (size_t)n*SCH + o]);
}

// ---------------------------------------------------------------------------
// feat = BN over batch of x upper-triangular pairs (thread per pair p)
// ---------------------------------------------------------------------------
__global__ void k_feat_bn(const float* __restrict__ x, const float* __restrict__ g,
                          const float* __restrict__ bb, float* __restrict__ feat)
{
    int p = blockIdx.x * blockDim.x + threadIdx.x;
    if (p >= TRIU_N) return;
    double disc = 799.0*799.0 - 8.0*(double)p;
    int i = (int)((799.0 - sqrt(disc)) * 0.5);
    if (i < 0) i = 0; if (i > 398) i = 398;
    while (i > 0 && (399*i - i*(i-1)/2) > p) --i;
    while ((399*(i+1) - (i+1)*i/2) <= p) ++i;
    int j = p - (399*i - i*(i-1)/2) + i + 1;
    float vals[16]; float m = 0.f;
    #pragma unroll
    for (int b = 0; b < 16; ++b) {
        vals[b] = x[((size_t)(b*F_DIM + i))*F_DIM + j]; m += vals[b];
    }
    m *= (1.f/16.f);
    float v = 0.f;
    #pragma unroll
    for (int b = 0; b < 16; ++b) { float d = vals[b]-m; v += d*d; }
    v *= (1.f/16.f);
    float inv = rsqrtf(v + 1e-5f);
    float gg = g[p], bo = bb[p];
    #pragma unroll
    for (int b = 0; b < 16; ++b)
        feat[(size_t)b*TRIU_N + p] = gg*(vals[b]-m)*inv + bo;
}

// graph-mean pooling of nf (batch is contiguous: node/400)
__global__ void k_gmean(const float* __restrict__ nf, float* __restrict__ gmr)
{
    int t = blockIdx.x * blockDim.x + threadIdx.x;
    if (t >= BGRAPH * 1024) return;
    int b = t >> 10, q = t & 1023;
    float s = 0.f;
    const float* base = nf + (size_t)b * F_DIM * 1024 + q;
    for (int n = 0; n < F_DIM; ++n) s += base[(size_t)n * 1024];
    gmr[t] = s * (1.f / (float)F_DIM);
}

// BN over 16 rows (+ optional relu). Per-column bias before BN cancels -> omitted upstream.
__global__ void k_bn_rows(const float* __restrict__ in, float* __restrict__ out,
                          const float* __restrict__ g, const float* __restrict__ bb,
                          int Q, int relu)
{
    int q = blockIdx.x * blockDim.x + threadIdx.x;
    if (q >= Q) return;
    float m = 0.f;
    for (int r = 0; r < 16; ++r) m += in[(size_t)r*Q + q];
    m *= (1.f/16.f);
    float v = 0.f;
    for (int r = 0; r < 16; ++r) { float d = in[(size_t)r*Q + q]-m; v += d*d; }
    v *= (1.f/16.f);
    float inv = rsqrtf(v + 1e-5f);
    float gg = g[q], bo = bb[q];
    for (int r = 0; r < 16; ++r) {
        float val = gg*(in[(size_t)r*Q + q]-m)*inv + bo;
        if (relu) val = fmaxf(val, 0.f);
        out[(size_t)r*Q + q] = val;
    }
}

// ---------------------------------------------------------------------------
// Split-K streaming GEMM for z@W1 (M=16, K=80824, N=512): bound by streaming
// 165 MB of W1 once. z chunk staged in LDS via async global->LDS B32 copies
// (per-lane piecewise feat/gmean addresses); W1 coalesced + prefetched.
// Partial sums atomically added into 32 KB L2-resident accumulator.
// ---------------------------------------------------------------------------
#define G1_KC 512
__global__ void k_gemm1(const float* __restrict__ feat, const float* __restrict__ gme,
                        const float* __restrict__ W1, float* __restrict__ out1)
{
    __shared__ __attribute__((aligned(16))) float zt[G1_KC * 16];
    const int kbase = blockIdx.x * G1_KC;
    const int kc = min(G1_KC, D1_N - kbase);
    for (int idx = threadIdx.x; idx < kc * 16; idx += blockDim.x) {
        int kk = idx >> 4, b = idx & 15;
        int k = kbase + kk;
        const float* src = (k < TRIU_N) ? (feat + (size_t)b*TRIU_N + k)
                                        : (gme + b*1024 + (k - TRIU_N));
        async_g2l_b32((uint32_t)(uintptr_t)&zt[kk*16 + b], src);
    }
    wait_async0();
    __syncthreads();
    const int hcol = blockIdx.y * 64 + (threadIdx.x & 63);
    const int ks   = threadIdx.x >> 6;
    float acc[16];
    #pragma unroll
    for (int b = 0; b < 16; ++b) acc[b] = 0.f;
    for (int kk = ks; kk < kc; kk += 4) {
        int kpf = kbase + kk + 128;
        if (kpf < D1_N) __builtin_prefetch(&W1[(size_t)kpf*HMID + hcol], 0, 0); // global_prefetch_b8
        float wv = W1[(size_t)(kbase + kk)*HMID + hcol];
        #pragma unroll
        for (int b = 0; b < 16; ++b) acc[b] += zt[kk*16 + b] * wv;
    }
    #pragma unroll
    for (int b = 0; b < 16; ++b) atomicAdd(&out1[b*HMID + hcol], acc[b]);
}

// final 256->7 head + log_softmax
__global__ void k_head(const float* __restrict__ z3, const float* __restrict__ W4,
                       const float* __restrict__ b4, float* __restrict__ out)
{
    __shared__ float lg[NCLS];
    int b = blockIdx.x, t = threadIdx.x;
    if (t < NCLS) {
        float s = b4[t];
        for (int k = 0; k < 256; ++k) s += z3[b*256 + k] * W4[k*NCLS + t];
        lg[t] = s;
    }
    __syncthreads();
    if (t == 0) {
        float mx = lg[0];
        for (int c = 1; c < NCLS; ++c) mx = fmaxf(mx, lg[c]);
        float se = 0.f;
        for (int c = 0; c < NCLS; ++c) se += expf(lg[c] - mx);
        float lse = mx + logf(se);
        for (int c = 0; c < NCLS; ++c) out[b*NCLS + c] = lg[c] - lse;
    }
}

// ---------------------------------------------------------------------------
extern "C" void kernel_launch(void* const* d_in, const int* in_sizes, int n_in,
                              void* d_out, int out_size, void* d_ws, size_t ws_size,
                              hipStream_t stream)
{
    (void)in_sizes; (void)n_in; (void)out_size;
    const float* x      = (const float*)d_in[0];
    const int*   eidx   = (const int*)  d_in[1];
    const float* evec   = (const float*)d_in[2];
    const float* eattr  = (const float*)d_in[3];
    /* d_in[4] = batch (unused: node/400) */
    const float* W_attr = (const float*)d_in[5];
    const float* b_attr = (const float*)d_in[6];
    const float* W_down = (const float*)d_in[7];
    const float* W_rad  = (const float*)d_in[8];
    const float* b_rad  = (const float*)d_in[9];
    const float* W_s    = (const float*)d_in[10];
    const float* b_s    = (const float*)d_in[11];
    const float* W_v    = (const float*)d_in[12];   // (2,64,256) == (128,256) row-major
    const float* W_sc   = (const float*)d_in[13];
    const float* bn_g   = (const float*)d_in[14];
    const float* bn_b   = (const float*)d_in[15];
    const float* bnh_g  = (const float*)d_in[16];
    const float* bnh_b  = (const float*)d_in[17];
    const float* W1     = (const float*)d_in[18];
    /* b1,b2,b3 cancel under BN mean subtraction */
    const float* g1     = (const float*)d_in[20];
    const float* be1    = (const float*)d_in[21];
    const float* W2     = (const float*)d_in[22];
    const float* g2     = (const float*)d_in[24];
    const float* be2    = (const float*)d_in[25];
    const float* W3     = (const float*)d_in[26];
    const float* g3     = (const float*)d_in[28];
    const float* be3    = (const float*)d_in[29];
    const float* W4     = (const float*)d_in[30];
    const float* b4     = (const float*)d_in[31];
    const int* src = eidx;
    const int* dst = eidx + NEDGE;

    // workspace layout (floats), with lifetime-based aliasing (~78 MB total)
    if (ws_size < (size_t)19467136 * sizeof(float)) return;
    float* ws   = (float*)d_ws;
    float* h1   = ws;                 // 2,560,000  (dead after h GEMM)
    float* h    = ws + 2560000;       //   409,600  (dead after k_edge)
    float* Y    = ws + 2969600;       // 3,276,800  (dead after k_edge)
    float* vecb = ws;                 // 4,915,200  (aliases h1/h/Y, written later)
    float* Abuf = ws + 6246400;       // 6,553,600  (dead after k_build_S)
    float* nf   = Abuf;               // alias (written after Abuf is dead)
    float* S    = ws + 12800000;      // 1,228,800
    float* V2   = ws + 14028800;      // 2,457,600
    float* scal = ws + 16486400;      // 1,638,400
    float* feat = ws + 18124800;      // 1,276,800
    float* gmr  = ws + 19401600;      // 16,384
    float* gme  = ws + 19417984;      // 16,384
    float* out1 = ws + 19434368;      // 8,192
    float* z1   = ws + 19442560;      // 8,192
    float* z2r  = ws + 19450752;      // 4,096
    float* z2   = ws + 19454848;      // 4,096
    float* z3r  = ws + 19458944;      // 4,096
    float* z3   = ws + 19463040;      // 4,096

    hipMemsetAsync(Abuf, 0, (size_t)NNODE * 1024 * sizeof(float), stream);
    hipMemsetAsync(out1, 0, (size_t)BGRAPH * HMID * sizeof(float), stream);

    const dim3 blk(128);
    // 1) h1 = relu(x @ W_attr + b_attr)        M=6400 K=400 N=400
    k_wmma_gemm<<<dim3(100, 25), blk, 0, stream>>>(x, W_attr, b_attr, h1, NNODE, 400, 400, 1);
    // 2) h = h1 @ W_down                        M=6400 K=400 N=64
    k_wmma_gemm<<<dim3(100, 4),  blk, 0, stream>>>(h1, W_down, nullptr, h, NNODE, 64, 400, 0);
    // 3) spherical harmonics
    k_sph<<<dim3((NEDGE + 255) / 256), dim3(256), 0, stream>>>(evec, Y);
    // 4) edge messages + scatter
    k_edge<<<dim3(NEDGE * CH / 256), dim3(256), 0, stream>>>(h, eattr, W_rad, b_rad, src, dst, Y, Abuf);
    // 5) node invariants
    k_build_S<<<dim3(NNODE * CH / 256), dim3(256), 0, stream>>>(Abuf, S, V2);
    // 6) scal = S @ W_s + b_s ; scal += x @ W_sc
    k_wmma_gemm<<<dim3(100, 16), blk, 0, stream>>>(S, W_s, b_s, scal, NNODE, SCH, 192, 0);
    k_wmma_gemm<<<dim3(100, 16), blk, 0, stream>>>(x, W_sc, nullptr, scal, NNODE, SCH, 400, 2);
    // 7) vec_i = V2_i @ W_v (128x256), i = 0..2
    for (int i = 0; i < 3; ++i)
        k_wmma_gemm<<<dim3(100, 16), blk, 0, stream>>>(V2 + (size_t)i*NNODE*128, W_v, nullptr,
                                                       vecb + (size_t)i*NNODE*SCH, NNODE, SCH, 128, 0);
    // 8) nf = tanh(concat)
    k_finalize_nf<<<dim3(NNODE * SCH / 256), dim3(256), 0, stream>>>(scal, vecb, nf);
    // 9) feat BN (upper-tri of x)
    k_feat_bn<<<dim3((TRIU_N + 255) / 256), dim3(256), 0, stream>>>(x, bn_g, bn_b, feat);
    // 10) graph mean + BN
    k_gmean<<<dim3(BGRAPH * 1024 / 256), dim3(256), 0, stream>>>(nf, gmr);
    k_bn_rows<<<dim3(4), dim3(256), 0, stream>>>(gmr, gme, bnh_g, bnh_b, 1024, 0);
    // 11) big streaming GEMM: out1 += z @ W1
    k_gemm1<<<dim3((D1_N + G1_KC - 1) / G1_KC, HMID / 64), dim3(256), 0, stream>>>(feat, gme, W1, out1);
    k_bn_rows<<<dim3(2), dim3(256), 0, stream>>>(out1, z1, g1, be1, HMID, 1);
    // 12) MLP tail (WMMA, M=16)
    k_wmma_gemm<<<dim3(1, 16), blk, 0, stream>>>(z1, W2, nullptr, z2r, BGRAPH, 256, 512, 0);
    k_bn_rows<<<dim3(1), dim3(256), 0, stream>>>(z2r, z2, g2, be2, 256, 1);
    k_wmma_gemm<<<dim3(1, 16), blk, 0, stream>>>(z2, W3, nullptr, z3r, BGRAPH, 256, 256, 0);
    k_bn_rows<<<dim3(1), dim3(256), 0, stream>>>(z3r, z3, g3, be3, 256, 1);
    // 13) head + log_softmax
    k_head<<<dim3(BGRAPH), dim3(32), 0, stream>>>(z3, W4, b4, (float*)d_out);
}